// MambaBlock_48180943126918
// MI455X (gfx1250) — compile-verified
//
#include <hip/hip_runtime.h>
#include <cmath>

typedef __attribute__((ext_vector_type(2))) float v2f;
typedef __attribute__((ext_vector_type(8))) float v8f;

#define D_MODEL_ 1024
#define D_STATE_ 16
#define D_CONV_  4
#define D_INNER_ 2048
#define DT_RANK_ 64
#define BATCH_   2
#define SEQ_     1024
#define NTOK_    (BATCH_ * SEQ_)
#define XDBL_LD_ 96              // dt_rank + 2*d_state
#define CHUNK_   64
#define NCHUNK_  (SEQ_ / CHUNK_) // 16
#define NBD_     (BATCH_ * D_INNER_) // 4096

__device__ __forceinline__ v8f wmma4(v2f a, v2f b, v8f c) {
  return __builtin_amdgcn_wmma_f32_16x16x4_f32(false, a, false, b, (short)0, c,
                                               false, false);
}

// ---------------------------------------------------------------------------
// LayerNorm: one block (256 threads) per token row of 1024 elements.
// ---------------------------------------------------------------------------
__global__ void ln_kernel(const float* __restrict__ x, const float* __restrict__ w,
                          const float* __restrict__ b, float* __restrict__ h) {
  const int row = blockIdx.x;
  const int tid = threadIdx.x;
  const float* xr = x + (size_t)row * D_MODEL_;
  float v0 = xr[tid], v1 = xr[tid + 256], v2 = xr[tid + 512], v3 = xr[tid + 768];
  __shared__ float ssum[256], ssq[256];
  ssum[tid] = v0 + v1 + v2 + v3;
  ssq[tid]  = v0*v0 + v1*v1 + v2*v2 + v3*v3;
  __syncthreads();
  for (int off = 128; off > 0; off >>= 1) {
    if (tid < off) { ssum[tid] += ssum[tid + off]; ssq[tid] += ssq[tid + off]; }
    __syncthreads();
  }
  const float mu  = ssum[0] * (1.0f / D_MODEL_);
  const float var = ssq[0] * (1.0f / D_MODEL_) - mu * mu;
  const float rs  = rsqrtf(var + 1e-5f);
  float* hr = h + (size_t)row * D_MODEL_;
  hr[tid]       = (v0 - mu) * rs * w[tid]       + b[tid];
  hr[tid + 256] = (v1 - mu) * rs * w[tid + 256] + b[tid + 256];
  hr[tid + 512] = (v2 - mu) * rs * w[tid + 512] + b[tid + 512];
  hr[tid + 768] = (v3 - mu) * rs * w[tid + 768] + b[tid + 768];
}

// ---------------------------------------------------------------------------
// Register-blocked fp32 WMMA GEMM: C[M,N] = A[M,K]*W[N,K]^T (+ epilogue).
// Wave tile: 32(M) x 64(N) = 2x4 subtiles of 16x16 -> 8 accumulators.
// Per K-step of 4: 6 fragment loads (2 A + 4 B), 8 WMMAs => ~10.7 FLOP/B.
// Block: 8 waves as 2(M) x 4(N) -> 64 x 256 per block. N % 256 == 0 required.
// EPI: 0 plain, 1 softplus(acc + bias[n]), 2 acc + residual[m,n]
// ---------------------------------------------------------------------------
template <int EPI>
__global__ void gemm_wmma_rb(const float* __restrict__ A, int lda,
                             const float* __restrict__ W, int ldw,
                             float* __restrict__ C, int ldc, int K,
                             const float* __restrict__ extra) {
  const int lane = threadIdx.x & 31;
  const int wave = threadIdx.x >> 5;
  const int half = lane >> 4;
  const int lm   = lane & 15;
  const int wm   = wave >> 2;   // 0..1
  const int wn   = wave & 3;    // 0..3
  const int m0 = blockIdx.y * 64 + wm * 32;
  const int n0 = blockIdx.x * 256 + wn * 64;

  const float* Ap0 = A + (size_t)(m0 + lm) * lda + half * 2;
  const float* Ap1 = Ap0 + (size_t)16 * lda;
  const float* Wp0 = W + (size_t)(n0 + lm) * ldw + half * 2;
  const float* Wp1 = Wp0 + (size_t)16 * ldw;
  const float* Wp2 = Wp0 + (size_t)32 * ldw;
  const float* Wp3 = Wp0 + (size_t)48 * ldw;

  v8f acc[2][4] = {};
  for (int k = 0; k < K; k += 16) {
    __builtin_prefetch(Ap0 + k + 256, 0, 1);
    __builtin_prefetch(Ap1 + k + 256, 0, 1);
    __builtin_prefetch(Wp0 + k + 256, 0, 1);
    __builtin_prefetch(Wp1 + k + 256, 0, 1);
    __builtin_prefetch(Wp2 + k + 256, 0, 1);
    __builtin_prefetch(Wp3 + k + 256, 0, 1);
#pragma unroll
    for (int kk = 0; kk < 16; kk += 4) {
      const v2f a0 = *(const v2f*)(Ap0 + k + kk);
      const v2f a1 = *(const v2f*)(Ap1 + k + kk);
      const v2f b0 = *(const v2f*)(Wp0 + k + kk);
      const v2f b1 = *(const v2f*)(Wp1 + k + kk);
      const v2f b2 = *(const v2f*)(Wp2 + k + kk);
      const v2f b3 = *(const v2f*)(Wp3 + k + kk);
      acc[0][0] = wmma4(a0, b0, acc[0][0]);
      acc[0][1] = wmma4(a0, b1, acc[0][1]);
      acc[0][2] = wmma4(a0, b2, acc[0][2]);
      acc[0][3] = wmma4(a0, b3, acc[0][3]);
      acc[1][0] = wmma4(a1, b0, acc[1][0]);
      acc[1][1] = wmma4(a1, b1, acc[1][1]);
      acc[1][2] = wmma4(a1, b2, acc[1][2]);
      acc[1][3] = wmma4(a1, b3, acc[1][3]);
    }
  }
#pragma unroll
  for (int i = 0; i < 2; ++i) {
#pragma unroll
    for (int j = 0; j < 4; ++j) {
      const int col = n0 + j * 16 + lm;
      const int rowbase = m0 + i * 16 + half * 8;
#pragma unroll
      for (int v = 0; v < 8; ++v) {
        const int row = rowbase + v;
        float val = acc[i][j][v];
        if (EPI == 1) {
          val += extra[col];
          val = (val > 20.0f) ? val : log1pf(expf(val));
        } else if (EPI == 2) {
          val += extra[(size_t)row * ldc + col];
        }
        C[(size_t)row * ldc + col] = val;
      }
    }
  }
}

// ---------------------------------------------------------------------------
// Small-N WMMA GEMM (one 16x16 tile per wave) for x_proj (N=96).
// ---------------------------------------------------------------------------
__global__ void gemm_wmma_small(const float* __restrict__ A, int lda,
                                const float* __restrict__ W, int ldw,
                                float* __restrict__ C, int ldc, int K) {
  const int lane = threadIdx.x & 31;
  const int wave = threadIdx.x >> 5;
  const int half = lane >> 4;
  const int lm   = lane & 15;
  const int m0 = blockIdx.y * 128 + wave * 16;
  const int n0 = blockIdx.x * 16;
  const float* Ap = A + (size_t)(m0 + lm) * lda + half * 2;
  const float* Wp = W + (size_t)(n0 + lm) * ldw + half * 2;
  v8f acc = {};
  for (int k = 0; k < K; k += 16) {
    __builtin_prefetch(Ap + k + 256, 0, 1);
    __builtin_prefetch(Wp + k + 256, 0, 1);
#pragma unroll
    for (int kk = 0; kk < 16; kk += 4) {
      const v2f a = *(const v2f*)(Ap + k + kk);
      const v2f b = *(const v2f*)(Wp + k + kk);
      acc = wmma4(a, b, acc);
    }
  }
  const int col = n0 + lm;
  const int rowbase = m0 + half * 8;
#pragma unroll
  for (int v = 0; v < 8; ++v)
    C[(size_t)(rowbase + v) * ldc + col] = acc[v];
}

// ---------------------------------------------------------------------------
// Depthwise causal conv1d (k=4) + bias + SiLU. One thread per (token, channel).
// ---------------------------------------------------------------------------
__global__ void conv_silu_kernel(const float* __restrict__ xz, const float* __restrict__ cw,
                                 const float* __restrict__ cb, float* __restrict__ xc) {
  const int idx   = blockIdx.x * 256 + threadIdx.x;
  const int d     = idx & (D_INNER_ - 1);
  const int token = idx >> 11;
  const int l     = token & (SEQ_ - 1);
  const int base  = token - l;
  float acc = cb[d];
#pragma unroll
  for (int t = 0; t < D_CONV_; ++t) {
    const int ll = l - (D_CONV_ - 1) + t;
    if (ll >= 0)
      acc += xz[(size_t)(base + ll) * (2 * D_INNER_) + d] * cw[d * D_CONV_ + t];
  }
  xc[(size_t)token * D_INNER_ + d] = acc / (1.0f + expf(-acc));
}

// ---------------------------------------------------------------------------
// Chunked selective scan (exact two-level scan over L).
// Phase 1: per (b,d,chunk) -> chunk-local scan S (zero start) and sum(dt).
//          Decay product over chunk = exp(A_s * sum_dt).
// ---------------------------------------------------------------------------
__global__ void scan_phase1(const float* __restrict__ dtb, const float* __restrict__ xdbl,
                            const float* __restrict__ xc, const float* __restrict__ A_log,
                            float* __restrict__ Ssum, float* __restrict__ dtsum) {
  const int idx = blockIdx.x * 256 + threadIdx.x;   // < NBD_*NCHUNK_
  const int bd  = idx & (NBD_ - 1);                 // lanes: consecutive d
  const int c   = idx >> 12;                        // chunk in high bits
  const int d   = bd & (D_INNER_ - 1);
  const int b   = bd >> 11;
  float Arow[D_STATE_];
#pragma unroll
  for (int s = 0; s < D_STATE_; ++s) Arow[s] = -expf(A_log[d * D_STATE_ + s]);
  float S[D_STATE_];
#pragma unroll
  for (int s = 0; s < D_STATE_; ++s) S[s] = 0.0f;
  float sd = 0.0f;
  const int t0 = c * CHUNK_;
  for (int t = 0; t < CHUNK_; ++t) {
    const int token = b * SEQ_ + t0 + t;
    const float dtv = dtb[(size_t)token * D_INNER_ + d];
    const float xcv = xc[(size_t)token * D_INNER_ + d];
    const float* bc = xdbl + (size_t)token * XDBL_LD_;
    sd += dtv;
#pragma unroll
    for (int s = 0; s < D_STATE_; ++s) {
      const float a = expf(dtv * Arow[s]);
      S[s] = S[s] * a + dtv * bc[DT_RANK_ + s] * xcv;
    }
  }
  float* Sp = Ssum + ((size_t)bd * NCHUNK_ + c) * D_STATE_;
#pragma unroll
  for (int s = 0; s < D_STATE_; ++s) Sp[s] = S[s];
  dtsum[bd * NCHUNK_ + c] = sd;
}

// Phase 2: per (b,d) -> 16-step sequential combine; writes each chunk's true
// start state.
__global__ void scan_phase2(const float* __restrict__ Ssum, const float* __restrict__ dtsum,
                            const float* __restrict__ A_log, float* __restrict__ start) {
  const int bd = blockIdx.x * 256 + threadIdx.x;    // < NBD_
  const int d  = bd & (D_INNER_ - 1);
  float Arow[D_STATE_];
#pragma unroll
  for (int s = 0; s < D_STATE_; ++s) Arow[s] = -expf(A_log[d * D_STATE_ + s]);
  float h[D_STATE_];
#pragma unroll
  for (int s = 0; s < D_STATE_; ++s) h[s] = 0.0f;
  for (int c = 0; c < NCHUNK_; ++c) {
    const size_t off = ((size_t)bd * NCHUNK_ + c) * D_STATE_;
    float* st = start + off;
    const float* Sp = Ssum + off;
    const float sd = dtsum[bd * NCHUNK_ + c];
#pragma unroll
    for (int s = 0; s < D_STATE_; ++s) {
      st[s] = h[s];
      const float P = expf(sd * Arow[s]);
      h[s] = P * h[s] + Sp[s];
    }
  }
}

// Phase 3: per (b,d,chunk) -> re-run chunk from its true start state; fused
// y = (scan + D*x_conv) * silu(z).
__global__ void scan_phase3(const float* __restrict__ dtb, const float* __restrict__ xdbl,
                            const float* __restrict__ xc, const float* __restrict__ xz,
                            const float* __restrict__ A_log, const float* __restrict__ Dp,
                            const float* __restrict__ start, float* __restrict__ yout) {
  const int idx = blockIdx.x * 256 + threadIdx.x;   // < NBD_*NCHUNK_
  const int bd  = idx & (NBD_ - 1);
  const int c   = idx >> 12;
  const int d   = bd & (D_INNER_ - 1);
  const int b   = bd >> 11;
  float Arow[D_STATE_];
#pragma unroll
  for (int s = 0; s < D_STATE_; ++s) Arow[s] = -expf(A_log[d * D_STATE_ + s]);
  const float Dd = Dp[d];
  float h[D_STATE_];
  const float* st = start + ((size_t)bd * NCHUNK_ + c) * D_STATE_;
#pragma unroll
  for (int s = 0; s < D_STATE_; ++s) h[s] = st[s];
  const int t0 = c * CHUNK_;
  for (int t = 0; t < CHUNK_; ++t) {
    const int token = b * SEQ_ + t0 + t;
    const float dtv = dtb[(size_t)token * D_INNER_ + d];
    const float xcv = xc[(size_t)token * D_INNER_ + d];
    const float zv  = xz[(size_t)token * (2 * D_INNER_) + D_INNER_ + d];
    const float* bc = xdbl + (size_t)token * XDBL_LD_;
    float y = 0.0f;
#pragma unroll
    for (int s = 0; s < D_STATE_; ++s) {
      const float a = expf(dtv * Arow[s]);
      h[s] = h[s] * a + dtv * bc[DT_RANK_ + s] * xcv;
      y += h[s] * bc[DT_RANK_ + D_STATE_ + s];
    }
    y += xcv * Dd;
    const float sil = zv / (1.0f + expf(-zv));
    yout[(size_t)token * D_INNER_ + d] = y * sil;
  }
}

// ---------------------------------------------------------------------------
extern "C" void kernel_launch(void* const* d_in, const int* in_sizes, int n_in,
                              void* d_out, int out_size, void* d_ws, size_t ws_size,
                              hipStream_t stream) {
  const float* x         = (const float*)d_in[0];
  const float* ln_w      = (const float*)d_in[1];
  const float* ln_b      = (const float*)d_in[2];
  const float* in_proj_w = (const float*)d_in[3];   // (4096, 1024)
  const float* conv_w    = (const float*)d_in[4];   // (2048, 1, 4)
  const float* conv_b    = (const float*)d_in[5];
  const float* x_proj_w  = (const float*)d_in[6];   // (96, 2048)
  const float* dt_proj_w = (const float*)d_in[7];   // (2048, 64)
  const float* dt_proj_b = (const float*)d_in[8];
  const float* A_log     = (const float*)d_in[9];   // (2048, 16)
  const float* Dp        = (const float*)d_in[10];
  const float* out_proj_w= (const float*)d_in[11];  // (1024, 2048)
  float* out = (float*)d_out;

  float* ws    = (float*)d_ws;
  float* h     = ws;                                    // 2048*1024
  float* xz    = h     + (size_t)NTOK_ * D_MODEL_;      // 2048*4096
  float* xconv = xz    + (size_t)NTOK_ * 2 * D_INNER_;  // 2048*2048
  float* xdbl  = xconv + (size_t)NTOK_ * D_INNER_;      // 2048*96
  float* dtb   = xdbl  + (size_t)NTOK_ * XDBL_LD_;      // 2048*2048
  float* yb    = dtb   + (size_t)NTOK_ * D_INNER_;      // 2048*2048
  float* Ssum  = yb    + (size_t)NTOK_ * D_INNER_;      // 4096*16*16
  float* strt  = Ssum  + (size_t)NBD_ * NCHUNK_ * D_STATE_; // 4096*16*16
  float* dtsum = strt  + (size_t)NBD_ * NCHUNK_ * D_STATE_; // 4096*16

  // 1) LayerNorm
  ln_kernel<<<NTOK_, 256, 0, stream>>>(x, ln_w, ln_b, h);

  // 2) in_proj: xz[2048,4096] = h * in_proj_w^T
  gemm_wmma_rb<0><<<dim3(2 * D_INNER_ / 256, NTOK_ / 64), 256, 0, stream>>>(
      h, D_MODEL_, in_proj_w, D_MODEL_, xz, 2 * D_INNER_, D_MODEL_, nullptr);

  // 3) depthwise causal conv + SiLU
  conv_silu_kernel<<<(NTOK_ * D_INNER_) / 256, 256, 0, stream>>>(xz, conv_w, conv_b, xconv);

  // 4) x_proj: xdbl[2048,96] = xconv * x_proj_w^T  (small N path)
  gemm_wmma_small<<<dim3(XDBL_LD_ / 16, NTOK_ / 128), 256, 0, stream>>>(
      xconv, D_INNER_, x_proj_w, D_INNER_, xdbl, XDBL_LD_, D_INNER_);

  // 5) dt_proj (+bias, softplus): dtb[2048,2048] = xdbl[:, :64] * dt_proj_w^T
  gemm_wmma_rb<1><<<dim3(D_INNER_ / 256, NTOK_ / 64), 256, 0, stream>>>(
      xdbl, XDBL_LD_, dt_proj_w, DT_RANK_, dtb, D_INNER_, DT_RANK_, dt_proj_b);

  // 6) chunked selective scan
  scan_phase1<<<(NBD_ * NCHUNK_) / 256, 256, 0, stream>>>(dtb, xdbl, xconv, A_log,
                                                          Ssum, dtsum);
  scan_phase2<<<NBD_ / 256, 256, 0, stream>>>(Ssum, dtsum, A_log, strt);
  scan_phase3<<<(NBD_ * NCHUNK_) / 256, 256, 0, stream>>>(dtb, xdbl, xconv, xz,
                                                          A_log, Dp, strt, yb);

  // 7) out_proj + residual: out = yb * out_proj_w^T + x
  gemm_wmma_rb<2><<<dim3(D_MODEL_ / 256, NTOK_ / 64), 256, 0, stream>>>(
      yb, D_INNER_, out_proj_w, D_INNER_, out, D_MODEL_, D_INNER_, x);
}